// LinearOperator2d_2929167695903
// MI455X (gfx1250) — compile-verified
//
#include <hip/hip_runtime.h>
#include <hip/hip_bf16.h>

typedef __attribute__((ext_vector_type(2))) float v2f;
typedef __attribute__((ext_vector_type(8))) float v8f;

#define HW 28
#define HID 64
#define BN 8
#define ON 16
#define IN 16
// padded LDS row strides (floats) to avoid 64-bank conflicts
#define HS_STRIDE 68
#define K2_STRIDE 260
#define V_STRIDE  29

__global__ __launch_bounds__(256) void linop2d_kernel(
    const float* __restrict__ v,     // (8,16,28,28)
    const float* __restrict__ W1,    // (4,64)
    const float* __restrict__ b1,    // (64)
    const float* __restrict__ W2,    // (64,256)
    const float* __restrict__ b2,    // (256)
    const float* __restrict__ bias,  // (16,1,1)
    float* __restrict__ out)         // (8,16,28,28)
{
    __shared__ float sHsum[32 * HS_STRIDE];   // Hsum[p][h], rows 28..31 zero-padded
    __shared__ float sK2[32 * K2_STRIDE];     // K2[p][o*16+i]
    __shared__ float sV[BN * IN * V_STRIDE];  // v[n,i,p, s=fixed]

    const int tid = threadIdx.x;
    const int rs  = blockIdx.x;
    const int r   = rs / HW;
    const int s   = rs % HW;
    const float inv = 1.0f / (float)HW;
    const float cr  = ((float)r + 0.5f) * inv;
    const float cs  = ((float)s + 0.5f) * inv;

    // Warm W2 (64 KB, reused by all 784 blocks) into cache hierarchy.
    for (int off = tid; off < 1024; off += 256)
        __builtin_prefetch(W2 + off * 16, 0, 3);

    // Stage v[:, :, :, s] into LDS (3584 floats).
    for (int idx = tid; idx < BN * IN * HW; idx += 256) {
        int ni = idx / HW;   // n*16+i
        int p  = idx % HW;
        sV[ni * V_STRIDE + p] = v[(ni * HW + p) * HW + s];
    }

    // ---- Phase A: Hsum[p][h] = sum_q gelu(c(p)W1[0,h] + c(q)W1[1,h] + cr*W1[2,h] + cs*W1[3,h] + b1[h])
    // 28*64 = 1792 (p,h) pairs; 7 per thread; 28-long q reduction each.
    for (int k = 0; k < 7; ++k) {
        int idx = tid + 256 * k;     // 0..1791
        int p = idx >> 6;            // 0..27
        int h = idx & 63;
        float w0 = W1[h];
        float w1 = W1[64 + h];
        float w2 = W1[128 + h];
        float w3 = W1[192 + h];
        float base = ((float)p + 0.5f) * inv * w0 + cr * w2 + cs * w3 + b1[h];
        float sum = 0.0f;
        #pragma unroll 4
        for (int q = 0; q < HW; ++q) {
            float x = base + ((float)q + 0.5f) * inv * w1;
            // exact GELU: 0.5*x*(1+erf(x/sqrt(2)))
            sum += 0.5f * x * (1.0f + erff(x * 0.70710678118654752f));
        }
        sHsum[p * HS_STRIDE + h] = sum;
    }
    // zero-pad rows 28..31 (256 entries, one per thread)
    {
        int p = 28 + (tid >> 6);
        int h = tid & 63;
        sHsum[p * HS_STRIDE + h] = 0.0f;
    }
    __syncthreads();

    // ---- Phase B: K2(32x256) = Hsum(32x64) @ W2(64x256) + 28*b2, via V_WMMA_F32_16X16X4_F32.
    // 2 M-tiles x 16 N-tiles = 32 output tiles; wave wv owns mtile = wv&1, ntiles (wv>>1)*4 .. +3.
    {
        const int wv    = tid >> 5;
        const int lane  = tid & 31;
        const int mtile = wv & 1;
        const int nbase = (wv >> 1) * 4;
        const int lrow  = lane & 15;
        const int khalf = lane >> 4;              // 0: K pair {0,1}, 1: K pair {2,3}
        const int m     = mtile * 16 + lrow;
        const float* hrow = &sHsum[m * HS_STRIDE];

        for (int nt = 0; nt < 4; ++nt) {
            const int ncol = (nbase + nt) * 16 + lrow;
            v8f c = {};
            #pragma unroll
            for (int ks = 0; ks < 16; ++ks) {
                const int k0 = ks * 4 + 2 * khalf;
                v2f a, b;
                // A 16x4 f32 layout: lanes 0-15 -> K={k0,k0+1}, lanes 16-31 -> K={k0+2,k0+3}
                a.x = hrow[k0];
                a.y = hrow[k0 + 1];
                // B 4x16 f32: row K striped across 16 lanes, lane-halves split the K rows
                b.x = W2[k0 * 256 + ncol];
                b.y = W2[(k0 + 1) * 256 + ncol];
                c = __builtin_amdgcn_wmma_f32_16x16x4_f32(
                        /*neg_a=*/false, a, /*neg_b=*/false, b,
                        /*c_mod=*/(short)0, c, /*reuse_a=*/false, /*reuse_b=*/false);
            }
            const float bb = 28.0f * b2[ncol];
            #pragma unroll
            for (int g = 0; g < 8; ++g) {
                // C/D layout: VGPR g -> M = g (lanes 0-15) or g+8 (lanes 16-31)
                int row = mtile * 16 + g + khalf * 8;
                sK2[row * K2_STRIDE + ncol] = c[g] + bb;
            }
        }
    }
    __syncthreads();

    // ---- Phase C: out[n,o,r,s] = quad * sum_{p,i} K2[p][o*16+i] * v[n,i,p,s] + bias[o]
    if (tid < BN * ON) {
        const int n = tid >> 4;
        const int o = tid & 15;
        float acc = 0.0f;
        for (int p = 0; p < HW; ++p) {
            const float* k2row = &sK2[p * K2_STRIDE + o * IN];
            #pragma unroll
            for (int i = 0; i < IN; ++i)
                acc += k2row[i] * sV[(n * IN + i) * V_STRIDE + p];
        }
        out[((n * ON + o) * HW + r) * HW + s] = acc * (inv * inv) + bias[o];
    }
}

extern "C" void kernel_launch(void* const* d_in, const int* in_sizes, int n_in,
                              void* d_out, int out_size, void* d_ws, size_t ws_size,
                              hipStream_t stream) {
    (void)in_sizes; (void)n_in; (void)out_size; (void)d_ws; (void)ws_size;
    const float* v    = (const float*)d_in[0];
    const float* W1   = (const float*)d_in[1];
    const float* b1   = (const float*)d_in[2];
    const float* W2   = (const float*)d_in[3];
    const float* b2   = (const float*)d_in[4];
    const float* bias = (const float*)d_in[5];
    float* out = (float*)d_out;

    linop2d_kernel<<<dim3(HW * HW), dim3(256), 0, stream>>>(
        v, W1, b1, W2, b2, bias, out);
}